// diffusion_73478300500629
// MI455X (gfx1250) — compile-verified
//
#include <hip/hip_runtime.h>
#include <hip/hip_bf16.h>

// ---------------------------------------------------------------------------
// Graph diffusion (label propagation), MI455X / gfx1250, wave32.
//   G = Z; G[mask] = Y;  w_e = deg(col)^-1/2 * deg(row)^-1/2
//   repeat 10x: x = LAM * (sum_e w_e * x[col_e] -> row_e) + (1-LAM) * G
//
// CSR by destination row, padded so every 16-row block's edge segment is a
// multiple of 4 records (16B-aligned pairs). Hot loop: branch-free segmented
// reduction through V_WMMA_F32_16X16X4_F32 (A = one-hot(localRow)*weight,
// B = gathered features, 3 column tiles for C=48). Edge record = 8 bytes:
// {(localRow<<27)|(col*48), weight} -> no multiply in the hot loop; one b128
// load fetches a lane's two records. Scalar (readfirstlane) loop bounds keep
// loop control on the SALU and EXEC all-ones for WMMA.
// Working set (~70 MB) lives in the 192 MB L2 -> L2-bandwidth bound.
// ---------------------------------------------------------------------------

typedef float v2f __attribute__((ext_vector_type(2)));
typedef float v8f __attribute__((ext_vector_type(8)));

#define CDIM 48
#define LAMBDA 0.9f
#define COLMASK 0x07FFFFFFu

// ---- setup kernels ---------------------------------------------------------

__global__ void deg_count_kernel(const int* __restrict__ ei,
                                 float* __restrict__ deg,
                                 unsigned* __restrict__ rcnt, int E) {
  int e = blockIdx.x * blockDim.x + threadIdx.x;
  if (e >= E) return;
  int r = ei[e];          // edge_index[0][e]
  int c = ei[E + e];      // edge_index[1][e]
  atomicAdd(&deg[c], 1.0f);
  atomicAdd(&rcnt[r], 1u);
}

__global__ void dinv_kernel(float* __restrict__ deg, int N) {
  int i = blockIdx.x * blockDim.x + threadIdx.x;
  if (i < N) deg[i] = 1.0f / sqrtf(deg[i]);   // deg >= 1 guaranteed
}

// Single-workgroup scan over 16-row blocks. Produces:
//   bptr[g]   : padded start of block g's edge segment (multiple of 4)
//   bend[g]   : end of real edges in block g (pad region = [bend, bptr[g+1]))
//   cursor[r] : placement cursor (start of row r's segment)
__global__ void scan_kernel(const unsigned* __restrict__ rcnt,
                            unsigned* __restrict__ cursor,
                            unsigned* __restrict__ bptr,
                            unsigned* __restrict__ bend,
                            int N, int nblocks) {
  __shared__ unsigned sums[1024];
  int tid = threadIdx.x;
  int chunkB = (nblocks + 1023) >> 10;
  int gs = tid * chunkB;
  int ge = gs + chunkB;
  if (ge > nblocks) ge = nblocks;
  if (gs > nblocks) gs = nblocks;

  unsigned local = 0;
  for (int g = gs; g < ge; ++g) {
    unsigned bc = 0;
    int r0 = g << 4;
    int r1 = r0 + 16; if (r1 > N) r1 = N;
    for (int r = r0; r < r1; ++r) bc += rcnt[r];
    local += (bc + 3u) & ~3u;                    // padded block size
  }
  sums[tid] = local;
  __syncthreads();
  for (int off = 1; off < 1024; off <<= 1) {     // Hillis-Steele inclusive
    unsigned t = (tid >= off) ? sums[tid - off] : 0u;
    __syncthreads();
    sums[tid] += t;
    __syncthreads();
  }
  unsigned run = sums[tid] - local;              // exclusive base of my chunk
  for (int g = gs; g < ge; ++g) {
    bptr[g] = run;
    int r0 = g << 4;
    int r1 = r0 + 16; if (r1 > N) r1 = N;
    unsigned pos = run, bc = 0;
    for (int r = r0; r < r1; ++r) {
      cursor[r] = pos;
      unsigned c = rcnt[r];
      pos += c; bc += c;
    }
    bend[g] = run + bc;
    run += (bc + 3u) & ~3u;
  }
  if (tid == 1023) bptr[nblocks] = sums[1023];
}

// Place each edge as an 8-byte record {(localRow<<27)|(col*CDIM), weight}.
// col*CDIM is pre-scaled so the hot loop needs no multiply.
__global__ void place_kernel(const int* __restrict__ ei,
                             const float* __restrict__ dinv,
                             unsigned* __restrict__ cursor,
                             uint2* __restrict__ rec, int E) {
  int e = blockIdx.x * blockDim.x + threadIdx.x;
  if (e >= E) return;
  int r = ei[e];
  int c = ei[E + e];
  unsigned pos = atomicAdd(&cursor[r], 1u);
  float w = dinv[r] * dinv[c];
  unsigned packed = ((unsigned)(r & 15) << 27) |
                    (((unsigned)c * CDIM) & COLMASK);
  rec[pos] = make_uint2(packed, __float_as_uint(w));
}

// Fill each block's pad region with {local=0, offset=0, w=0} records.
__global__ void pad_fill_kernel(const unsigned* __restrict__ bptr,
                                const unsigned* __restrict__ bend,
                                uint2* __restrict__ rec, int nblocks) {
  int g = blockIdx.x * blockDim.x + threadIdx.x;
  if (g >= nblocks) return;
  unsigned s = bend[g], e = bptr[g + 1];
  for (unsigned p = s; p < e; ++p) rec[p] = make_uint2(0u, 0u);
}

__global__ void copy4_kernel(const float4* __restrict__ src,
                             float4* __restrict__ dst, int n4) {
  int i = blockIdx.x * blockDim.x + threadIdx.x;
  if (i < n4) dst[i] = src[i];
}

__global__ void mask_scatter_kernel(const float* __restrict__ Y,
                                    const int* __restrict__ mask,
                                    float* __restrict__ G, int M) {
  int i = blockIdx.x * blockDim.x + threadIdx.x;
  if (i >= M * CDIM) return;
  int m = i / CDIM;
  int c = i - m * CDIM;
  G[mask[m] * CDIM + c] = Y[i];
}

// ---- main diffusion step: padded-CSR segmented reduction via WMMA ---------
// One wave owns 16 destination rows; each k-step consumes 4 edges.
// A(16x4): A[m][k] = w_k iff localRow_k == m. B(4x16): gathered features.
__global__ __launch_bounds__(256)
void diffusion_step_kernel(const float* __restrict__ x,
                           const float* __restrict__ G,
                           float* __restrict__ out,
                           const unsigned* __restrict__ bptr,
                           const uint2* __restrict__ rec,
                           int nblocks, int N) {
  int wave = threadIdx.x >> 5;
  int lane = threadIdx.x & 31;
  int group = blockIdx.x * (blockDim.x >> 5) + wave;
  if (group >= nblocks) return;                 // wave-uniform: EXEC stays full

  int base = group << 4;                        // first of 16 rows
  int half = lane >> 4;                         // lanes 16-31 handle K=2,3
  int m    = lane & 15;                         // A row / B column in tile

  // Wave-uniform loop bounds -> scalar loop control, EXEC stays all-ones.
  int js = __builtin_amdgcn_readfirstlane((int)bptr[group]);
  int je = __builtin_amdgcn_readfirstlane((int)bptr[group + 1]);

  v8f acc0 = {0.f,0.f,0.f,0.f,0.f,0.f,0.f,0.f};
  v8f acc1 = {0.f,0.f,0.f,0.f,0.f,0.f,0.f,0.f};
  v8f acc2 = {0.f,0.f,0.f,0.f,0.f,0.f,0.f,0.f};

#pragma unroll 2
  for (int j = js; j < je; j += 4) {
    // Lane's two records: edges j+2*half, j+2*half+1 (16B aligned pair).
    const uint4 rr = *(const uint4*)(rec + (j + 2 * half));
    unsigned p0 = rr.x, p1 = rr.z;
    float    w0 = __uint_as_float(rr.y);
    float    w1 = __uint_as_float(rr.w);
    int l0 = (int)(p0 >> 27);
    int l1 = (int)(p1 >> 27);
    int o0 = (int)(p0 & COLMASK) + m;           // col*CDIM pre-scaled
    int o1 = (int)(p1 & COLMASK) + m;

    v2f a;                                      // one-hot row select * weight
    a.x = (l0 == m) ? w0 : 0.0f;
    a.y = (l1 == m) ? w1 : 0.0f;

    v2f b0; b0.x = x[o0];      b0.y = x[o1];
    v2f b1; b1.x = x[o0 + 16]; b1.y = x[o1 + 16];
    v2f b2; b2.x = x[o0 + 32]; b2.y = x[o1 + 32];

    acc0 = __builtin_amdgcn_wmma_f32_16x16x4_f32(false, a, false, b0,
                                                 (short)0, acc0, false, false);
    acc1 = __builtin_amdgcn_wmma_f32_16x16x4_f32(false, a, false, b1,
                                                 (short)0, acc1, false, false);
    acc2 = __builtin_amdgcn_wmma_f32_16x16x4_f32(false, a, false, b2,
                                                 (short)0, acc2, false, false);
  }

  // D layout: VGPR v -> row base + v + 8*half; column = m (+16 per tile).
  int row0 = base + 8 * half;
  const float omL = 1.0f - LAMBDA;
  if (base + 16 <= N) {                         // uniform: full block
#pragma unroll
    for (int v = 0; v < 8; ++v) {
      int o = (row0 + v) * CDIM + m;
      out[o]      = fmaf(LAMBDA, acc0[v], omL * G[o]);
      out[o + 16] = fmaf(LAMBDA, acc1[v], omL * G[o + 16]);
      out[o + 32] = fmaf(LAMBDA, acc2[v], omL * G[o + 32]);
    }
  } else {                                      // tail block only
#pragma unroll
    for (int v = 0; v < 8; ++v) {
      int row = row0 + v;
      if (row < N) {
        int o = row * CDIM + m;
        out[o]      = fmaf(LAMBDA, acc0[v], omL * G[o]);
        out[o + 16] = fmaf(LAMBDA, acc1[v], omL * G[o + 16]);
        out[o + 32] = fmaf(LAMBDA, acc2[v], omL * G[o + 32]);
      }
    }
  }
}

// ---- host orchestration ----------------------------------------------------

extern "C" void kernel_launch(void* const* d_in, const int* in_sizes, int n_in,
                              void* d_out, int out_size, void* d_ws, size_t ws_size,
                              hipStream_t stream) {
  const float* Z    = (const float*)d_in[0];
  const float* Y    = (const float*)d_in[1];
  const int*   mask = (const int*)d_in[2];
  const int*   ei   = (const int*)d_in[3];
  // d_in[4] (beta) is unused by the reference.

  int N = in_sizes[0] / CDIM;
  int M = in_sizes[1] / CDIM;
  int E = in_sizes[3] / 2;
  int nblocks = (N + 15) / 16;

  char* p = (char*)d_ws;
  auto alloc = [&](size_t bytes) -> char* {
    char* r = p;
    p += (bytes + 255) & ~(size_t)255;
    return r;
  };
  float*    deg    = (float*)   alloc((size_t)N * 4);       // becomes dinv
  unsigned* rcnt   = (unsigned*)alloc((size_t)N * 4);
  unsigned* cursor = (unsigned*)alloc((size_t)N * 4);
  unsigned* bptr   = (unsigned*)alloc((size_t)(nblocks + 1) * 4);
  unsigned* bend   = (unsigned*)alloc((size_t)nblocks * 4);
  uint2*    rec    = (uint2*)   alloc(((size_t)E + 4u * nblocks) * 8);
  float*    G      = (float*)   alloc((size_t)N * CDIM * 4);
  float*    bufA   = (float*)   alloc((size_t)N * CDIM * 4);
  float*    bufB   = (float*)   alloc((size_t)N * CDIM * 4);

  hipMemsetAsync(deg,  0, (size_t)N * 4, stream);
  hipMemsetAsync(rcnt, 0, (size_t)N * 4, stream);

  const int tb = 256;
  deg_count_kernel<<<(E + tb - 1) / tb, tb, 0, stream>>>(ei, deg, rcnt, E);
  dinv_kernel<<<(N + tb - 1) / tb, tb, 0, stream>>>(deg, N);
  scan_kernel<<<1, 1024, 0, stream>>>(rcnt, cursor, bptr, bend, N, nblocks);
  place_kernel<<<(E + tb - 1) / tb, tb, 0, stream>>>(ei, deg, cursor, rec, E);
  pad_fill_kernel<<<(nblocks + tb - 1) / tb, tb, 0, stream>>>(bptr, bend, rec,
                                                              nblocks);
  int nc4 = (N * CDIM) / 4;
  copy4_kernel<<<(nc4 + tb - 1) / tb, tb, 0, stream>>>((const float4*)Z,
                                                       (float4*)G, nc4);
  mask_scatter_kernel<<<(M * CDIM + tb - 1) / tb, tb, 0, stream>>>(Y, mask, G, M);

  const int waves_per_blk = 8;                  // 256 threads = 8 wave32
  int blks = (nblocks + waves_per_blk - 1) / waves_per_blk;

  const float* src = G;                         // x starts as G
  for (int it = 0; it < 10; ++it) {
    float* dst = (it == 9) ? (float*)d_out : ((it & 1) ? bufB : bufA);
    diffusion_step_kernel<<<blks, 32 * waves_per_blk, 0, stream>>>(
        src, G, dst, bptr, rec, nblocks, N);
    src = dst;
  }
}